// SetConv_11802570130411
// MI455X (gfx1250) — compile-verified
//
#include <hip/hip_runtime.h>
#include <hip/hip_bf16.h>

typedef __attribute__((ext_vector_type(2))) float v2f;
typedef __attribute__((ext_vector_type(8))) float v8f;

constexpr int BB  = 4;
constexpr int NN  = 8192;
constexpr int CIN = 64;
constexpr int MM  = 2048;
constexpr int SS  = 32;
constexpr float R2  = 0.25f;     // RADIUS^2
constexpr float EPS = 1e-3f;

// ---------------------------------------------------------------------------
// gfx1250 async global->LDS copies (ASYNCcnt path). VDST carries the LDS byte
// address (low 32 bits of a generic LDS pointer), VADDR the 64-bit global
// address (GV mode, saddr=off). Drained with s_wait_asynccnt before barrier.
// ---------------------------------------------------------------------------
__device__ __forceinline__ void async_copy_b128(unsigned lds_addr, const float* g) {
  asm volatile("global_load_async_to_lds_b128 %0, %1, off"
               :: "v"(lds_addr), "v"(g) : "memory");
}
__device__ __forceinline__ void async_copy_b32(unsigned lds_addr, const float* g) {
  asm volatile("global_load_async_to_lds_b32 %0, %1, off"
               :: "v"(lds_addr), "v"(g) : "memory");
}
__device__ __forceinline__ void wait_async0() {
  asm volatile("s_wait_asynccnt 0" ::: "memory");
}

// ---------------------------------------------------------------------------
// Kernel 1: farthest point sampling (one workgroup per batch, 256 threads).
// Points staged in LDS; per-step argmax = wave shuffle reduce + tiny LDS pass.
// Writes new_points (B,3,M) directly into d_out[0 : B*3*M].
// ---------------------------------------------------------------------------
__global__ void fps_kernel(const float* __restrict__ points,
                           float* __restrict__ out_np) {
  extern __shared__ float sm[];
  float* px = sm;
  float* py = px + NN;
  float* pz = py + NN;
  float* rv = pz + NN;        // 8 per-wave max values
  int*   ri = (int*)(rv + 8); // 8 per-wave argmax indices
  int*   bc = ri + 8;         // broadcast slot

  const int t    = threadIdx.x;
  const int lane = t & 31;
  const int wid  = t >> 5;
  const int b    = blockIdx.x;
  const float* P = points + (size_t)b * 3 * NN;

  for (int i = t; i < NN; i += 256) {
    px[i] = P[i];
    py[i] = P[NN + i];
    pz[i] = P[2 * NN + i];
  }
  __syncthreads();

  float dists[32];
#pragma unroll
  for (int i = 0; i < 32; ++i) dists[i] = 1e10f;

  int cur = 0;
  for (int step = 0; step < MM; ++step) {
    float sx = px[cur], sy = py[cur], sz = pz[cur];
    if (t == 0) {
      out_np[(size_t)b * 3 * MM + step]          = sx;
      out_np[(size_t)b * 3 * MM + MM + step]     = sy;
      out_np[(size_t)b * 3 * MM + 2 * MM + step] = sz;
    }
    float mv = -1.0f; int mj = 0;
#pragma unroll
    for (int i = 0; i < 32; ++i) {
      int j = i * 256 + t;
      float dx = px[j] - sx, dy = py[j] - sy, dz = pz[j] - sz;
      float nd = fminf(dists[i], dx * dx + dy * dy + dz * dz);
      dists[i] = nd;
      if (nd > mv) { mv = nd; mj = j; }   // keeps first occurrence (lowest j)
    }
#pragma unroll
    for (int off = 16; off > 0; off >>= 1) {
      float ov = __shfl_xor(mv, off, 32);
      int   oi = __shfl_xor(mj, off, 32);
      if (ov > mv || (ov == mv && oi < mj)) { mv = ov; mj = oi; }
    }
    if (lane == 0) { rv[wid] = mv; ri[wid] = mj; }
    __syncthreads();
    if (t == 0) {
      float bv = rv[0]; int bi = ri[0];
      for (int k = 1; k < 8; ++k) {
        float ov = rv[k]; int oi = ri[k];
        if (ov > bv || (ov == bv && oi < bi)) { bv = ov; bi = oi; }
      }
      bc[0] = bi;
    }
    __syncthreads();
    cur = bc[0];
  }
}

// ---------------------------------------------------------------------------
// Kernel 2: ball query (one wave per query; ballot/popcount compaction).
// ---------------------------------------------------------------------------
__global__ void ballq_kernel(const float* __restrict__ points,
                             const float* __restrict__ newpts,
                             int* __restrict__ indb) {
  extern __shared__ float sm[];
  float* px = sm;
  float* py = px + NN;
  float* pz = py + NN;

  const int t    = threadIdx.x;
  const int lane = t & 31;
  const int wid  = t >> 5;
  const int bblk = blockIdx.x >> 9;
  const float* P = points + (size_t)bblk * 3 * NN;

  for (int i = t; i < NN; i += 128) {
    px[i] = P[i];
    py[i] = P[NN + i];
    pz[i] = P[2 * NN + i];
  }
  __syncthreads();

  const int q = blockIdx.x * 4 + wid;
  const int b = q >> 11;
  const int m = q & (MM - 1);

  const float qx = newpts[(size_t)b * 3 * MM + m];
  const float qy = newpts[(size_t)b * 3 * MM + MM + m];
  const float qz = newpts[(size_t)b * 3 * MM + 2 * MM + m];

  int cnt = 0, first = 0;
  bool found = false;
  for (int chunk = 0; chunk < NN / 32 && cnt < SS; ++chunk) {
    int j = chunk * 32 + lane;
    float dx = px[j] - qx, dy = py[j] - qy, dz = pz[j] - qz;
    bool inr = (dx * dx + dy * dy + dz * dz) <= R2;
    unsigned mask = (unsigned)__ballot(inr);
    if (!found && mask) { first = chunk * 32 + __ffs(mask) - 1; found = true; }
    if (inr) {
      int pos = cnt + __popc(mask & ((1u << lane) - 1u));
      if (pos < SS) indb[(size_t)q * SS + pos] = j;
    }
    cnt += __popc(mask);
  }
  if (cnt > SS) cnt = SS;
  int fillv = found ? first : 0;
  if (lane >= cnt) indb[(size_t)q * SS + lane] = fillv;
}

// ---------------------------------------------------------------------------
// Fused gather + 3x(conv1x1 + BN + ReLU) + maxpool using fp32 WMMA.
// ---------------------------------------------------------------------------
__device__ __forceinline__ void wmma_layer(const float* __restrict__ W, int KP,
                                           int COUT, const float* __restrict__ sc,
                                           const float* __restrict__ sh,
                                           const float* __restrict__ inb,
                                           float* __restrict__ outb,
                                           int lm, int hf) {
  for (int ct = 0; ct < COUT / 16; ++ct) {
    for (int s16 = 0; s16 < 2; ++s16) {
      v8f acc = {};
      for (int k = 0; k < KP; k += 4) {
        const float* wr = W + (ct * 16 + lm) * KP + k + 2 * hf;
        v2f a = { wr[0], wr[1] };
        const float* br = inb + (k + 2 * hf) * SS + s16 * 16 + lm;
        v2f bm = { br[0], br[SS] };
        acc = __builtin_amdgcn_wmma_f32_16x16x4_f32(
            false, a, false, bm, (short)0, acc, false, false);
      }
#pragma unroll
      for (int g = 0; g < 8; ++g) {
        int ch = ct * 16 + g + hf * 8;
        float y = acc[g] * sc[ch] + sh[ch];
        outb[ch * SS + s16 * 16 + lm] = fmaxf(y, 0.0f);
      }
    }
  }
}

__global__ void mlp_kernel(const float* __restrict__ points,
                           const float* __restrict__ features,
                           const float* __restrict__ w0, const float* __restrict__ b0,
                           const float* __restrict__ g0, const float* __restrict__ be0,
                           const float* __restrict__ mu0, const float* __restrict__ v0,
                           const float* __restrict__ w1, const float* __restrict__ b1,
                           const float* __restrict__ g1, const float* __restrict__ be1,
                           const float* __restrict__ mu1, const float* __restrict__ v1,
                           const float* __restrict__ w2, const float* __restrict__ b2,
                           const float* __restrict__ g2, const float* __restrict__ be2,
                           const float* __restrict__ mu2, const float* __restrict__ v2,
                           float* __restrict__ out, const int* __restrict__ ind) {
  extern __shared__ float sm[];
  float* w0s = sm;              // 64*68 (K padded 67->68, zero-filled)
  float* w1s = w0s + 64 * 68;   // 64*64  (16B-aligned offset)
  float* w2s = w1s + 64 * 64;   // 128*64 (16B-aligned offset)
  float* sc0 = w2s + 128 * 64;
  float* sh0 = sc0 + 64;
  float* sc1 = sh0 + 64;
  float* sh1 = sc1 + 64;
  float* sc2 = sh1 + 64;
  float* sh2 = sc2 + 128;
  float* bufA = sh2 + 128;            // 4 waves * 68*32
  float* bufB = bufA + 4 * 68 * 32;   // 4 waves * 128*32

  const int t    = threadIdx.x;
  const int lane = t & 31;
  const int wid  = t >> 5;
  const int lm   = lane & 15;
  const int hf   = lane >> 4;
  const int q    = blockIdx.x * 4 + wid;
  const int b    = q >> 11;
  const int m    = q & (MM - 1);

  // Async-stage the unpadded weight blocks straight into LDS (no VGPR bounce).
  for (int i = t * 4; i < 64 * 64; i += 128 * 4)
    async_copy_b128((unsigned)(size_t)(w1s + i), w1 + i);
  for (int i = t * 4; i < 128 * 64; i += 128 * 4)
    async_copy_b128((unsigned)(size_t)(w2s + i), w2 + i);

  // w0 needs the 67->68 zero-pad interleave: scalar path.
  for (int i = t; i < 64 * 68; i += 128) {
    int o = i / 68, c = i % 68;
    w0s[i] = (c < 67) ? w0[o * 67 + c] : 0.0f;
  }
  for (int i = t; i < 64; i += 128) {
    float s0 = g0[i] * rsqrtf(v0[i] + EPS);
    sc0[i] = s0; sh0[i] = s0 * (b0[i] - mu0[i]) + be0[i];
    float s1 = g1[i] * rsqrtf(v1[i] + EPS);
    sc1[i] = s1; sh1[i] = s1 * (b1[i] - mu1[i]) + be1[i];
  }
  for (int i = t; i < 128; i += 128) {
    float s2 = g2[i] * rsqrtf(v2[i] + EPS);
    sc2[i] = s2; sh2[i] = s2 * (b2[i] - mu2[i]) + be2[i];
  }

  // Gather x0 = [points[ind]-newpt ; features[ind]] into this wave's buffer.
  float* myA = bufA + wid * (68 * 32);
  float* myB = bufB + wid * (128 * 32);
  const int pj = ind[(size_t)q * SS + lane];
  const float* Pb = points + (size_t)b * 3 * NN;
  const float* Fb = features + (size_t)b * CIN * NN;
#pragma unroll
  for (int d = 0; d < 3; ++d) {
    float qd = out[(size_t)b * 3 * MM + d * MM + m];
    myA[d * SS + lane] = Pb[d * NN + pj] - qd;
  }
  // Scattered per-lane feature gather: global -> LDS async (per-lane VDST).
  for (int c = 0; c < CIN; ++c)
    async_copy_b32((unsigned)(size_t)(myA + (3 + c) * SS + lane),
                   Fb + c * NN + pj);
  myA[67 * SS + lane] = 0.0f;   // K padding row

  wait_async0();                 // drain ASYNCcnt before cross-wave barrier
  __syncthreads();

  wmma_layer(w0s, 68, 64,  sc0, sh0, myA, myB, lm, hf);
  __syncthreads();
  wmma_layer(w1s, 64, 64,  sc1, sh1, myB, myA, lm, hf);
  __syncthreads();
  wmma_layer(w2s, 64, 128, sc2, sh2, myA, myB, lm, hf);
  __syncthreads();

  // Maxpool over S=32; write pooled (B,128,M) after new_points in d_out.
#pragma unroll
  for (int r = 0; r < 4; ++r) {
    int c = lane + r * 32;
    const float* row = myB + c * SS;
    float mx = row[0];
#pragma unroll
    for (int s = 1; s < SS; ++s) mx = fmaxf(mx, row[s]);
    out[(size_t)BB * 3 * MM + ((size_t)b * 128 + c) * MM + m] = mx;
  }
}

// ---------------------------------------------------------------------------
extern "C" void kernel_launch(void* const* d_in, const int* in_sizes, int n_in,
                              void* d_out, int out_size, void* d_ws, size_t ws_size,
                              hipStream_t stream) {
  const float* points   = (const float*)d_in[0];
  const float* features = (const float*)d_in[1];
  const float* w0 = (const float*)d_in[2];
  const float* b0 = (const float*)d_in[3];
  const float* g0 = (const float*)d_in[4];
  const float* be0 = (const float*)d_in[5];
  const float* mu0 = (const float*)d_in[6];
  const float* v0 = (const float*)d_in[7];
  const float* w1 = (const float*)d_in[8];
  const float* b1 = (const float*)d_in[9];
  const float* g1 = (const float*)d_in[10];
  const float* be1 = (const float*)d_in[11];
  const float* mu1 = (const float*)d_in[12];
  const float* v1 = (const float*)d_in[13];
  const float* w2 = (const float*)d_in[14];
  const float* b2 = (const float*)d_in[15];
  const float* g2 = (const float*)d_in[16];
  const float* be2 = (const float*)d_in[17];
  const float* mu2 = (const float*)d_in[18];
  const float* v2 = (const float*)d_in[19];
  (void)in_sizes; (void)n_in; (void)out_size; (void)ws_size;

  float* out = (float*)d_out;
  int* ind = (int*)d_ws;   // B*M*S ints = 1 MiB

  size_t fps_lds = (size_t)(3 * NN + 8 + 8 + 1) * 4;
  fps_kernel<<<BB, 256, fps_lds, stream>>>(points, out);

  size_t bq_lds = (size_t)(3 * NN) * 4;
  ballq_kernel<<<(BB * MM) / 4, 128, bq_lds, stream>>>(points, out, ind);

  size_t mlp_lds = (size_t)(64 * 68 + 64 * 64 + 128 * 64 + 512 +
                            4 * 68 * 32 + 4 * 128 * 32) * 4;
  mlp_kernel<<<(BB * MM) / 4, 128, mlp_lds, stream>>>(
      points, features,
      w0, b0, g0, be0, mu0, v0,
      w1, b1, g1, be1, mu1, v1,
      w2, b2, g2, be2, mu2, v2,
      out, ind);
}